// GNN_10136122819050
// MI455X (gfx1250) — compile-verified
//
#include <hip/hip_runtime.h>

#define HD 128
#define LEAKY 0.1f
#define BN_EPS 1e-5f

typedef float v2f __attribute__((ext_vector_type(2)));
typedef float v8f __attribute__((ext_vector_type(8)));

// ---------------------------------------------------------------------------
// GEMM: Out[n,128] = act(A)[n,128] @ W[128,128]
// If ss != nullptr, A is transformed per-channel on load:
//   a = A*ss[k] + ss[128+k]; a = leakyrelu(a)
// One wave computes a 16(M) x 128(N) strip using v_wmma_f32_16x16x4_f32.
// ---------------------------------------------------------------------------
__global__ __launch_bounds__(128) void gcn_gemm_kernel(
    const float* __restrict__ A, const float* __restrict__ W,
    const float* __restrict__ ss, float* __restrict__ Out, int n)
{
  __shared__ float Wl[HD * HD];   // 64 KB: whole weight matrix
  int tid = threadIdx.x;
  // stage W into LDS (float4, fully coalesced)
  const float4* W4 = (const float4*)W;
  float4* Wl4 = (float4*)Wl;
#pragma unroll
  for (int i = 0; i < (HD * HD / 4) / 128; ++i)
    Wl4[tid + i * 128] = W4[tid + i * 128];
  __syncthreads();

  int wave = tid >> 5;
  int lane = tid & 31;
  int mtile = blockIdx.x * 4 + wave;          // 16-row tile index
  int m     = lane & 15;
  int rowA  = mtile * 16 + m;
  int koff  = (lane >> 4) << 1;               // 0 (lanes 0-15) or 2 (lanes 16-31)
  bool rowOk = rowA < n;
  const float* arow = A + (size_t)(rowOk ? rowA : 0) * HD;

  // A fragments for all K: afrag[kk] covers K = 4*kk+koff, +1  (ISA 16x4 f32 layout)
  v2f afrag[32];
#pragma unroll
  for (int kk = 0; kk < 32; ++kk) {
    int k0 = kk * 4 + koff;
    float x0 = rowOk ? arow[k0]     : 0.0f;
    float x1 = rowOk ? arow[k0 + 1] : 0.0f;
    if (ss != nullptr) {
      x0 = x0 * ss[k0]     + ss[HD + k0];
      x1 = x1 * ss[k0 + 1] + ss[HD + k0 + 1];
      x0 = x0 > 0.0f ? x0 : LEAKY * x0;
      x1 = x1 > 0.0f ? x1 : LEAKY * x1;
    }
    afrag[kk].x = x0;
    afrag[kk].y = x1;
  }

  int ncol = lane & 15;
  int rowD = mtile * 16 + (lane >> 4) * 8;    // D layout: lanes 16-31 hold M=8..15
  for (int nt = 0; nt < 8; ++nt) {
    v8f c = {};
#pragma unroll
    for (int kk = 0; kk < 32; ++kk) {
      int k0 = kk * 4 + koff;
      v2f b;  // B 4x16 layout mirrors A: VGPR0 = row k0, VGPR1 = row k0+1
      b.x = Wl[k0 * HD + nt * 16 + ncol];
      b.y = Wl[(k0 + 1) * HD + nt * 16 + ncol];
      c = __builtin_amdgcn_wmma_f32_16x16x4_f32(false, afrag[kk], false, b,
                                                (short)0, c, false, false);
    }
    int col = nt * 16 + ncol;
#pragma unroll
    for (int r = 0; r < 8; ++r) {
      int rr = rowD + r;
      if (rr < n) Out[(size_t)rr * HD + col] = c[r];
    }
  }
}

// ---------------------------------------------------------------------------
// Degree / rsqrt-degree (self loop folded in as +1)
// ---------------------------------------------------------------------------
__global__ void deg_kernel(const int* __restrict__ dst, float* __restrict__ deg, int nE)
{
  int e = blockIdx.x * blockDim.x + threadIdx.x;
  if (e < nE) unsafeAtomicAdd(&deg[dst[e]], 1.0f);
}

__global__ void dis_kernel(float* __restrict__ deg, int n)
{
  int i = blockIdx.x * blockDim.x + threadIdx.x;
  if (i < n) deg[i] = rsqrtf(deg[i] + 1.0f);   // deg >= 1 always (self loops)
}

__global__ void cnt_kernel(const int* __restrict__ batch, float* __restrict__ cnt, int n)
{
  int i = blockIdx.x * blockDim.x + threadIdx.x;
  if (i < n) unsafeAtomicAdd(&cnt[batch[i]], 1.0f);
}

// ---------------------------------------------------------------------------
// Edge aggregation: y[dst] += dis[src]*dis[dst] * h[src]
// One 32-lane group per edge; each lane handles 4 channels (float4).
// Edges e >= nE are the self loops (src = dst = e - nE).
// ---------------------------------------------------------------------------
__global__ __launch_bounds__(256) void agg_kernel(
    const float* __restrict__ h, const int* __restrict__ src,
    const int* __restrict__ dst, const float* __restrict__ dis,
    float* __restrict__ y, int nE, int nN)
{
  long long t = (long long)blockIdx.x * blockDim.x + threadIdx.x;
  int e    = (int)(t >> 5);
  int lane = (int)(t & 31);
  if (e >= nE + nN) return;
  int s, d;
  if (e < nE) { s = src[e]; d = dst[e]; } else { s = e - nE; d = s; }
  float norm = dis[s] * dis[d];
  float4 v = ((const float4*)(h + (size_t)s * HD))[lane];
  float* yp = y + (size_t)d * HD + lane * 4;
  unsafeAtomicAdd(yp + 0, norm * v.x);
  unsafeAtomicAdd(yp + 1, norm * v.y);
  unsafeAtomicAdd(yp + 2, norm * v.z);
  unsafeAtomicAdd(yp + 3, norm * v.w);
}

// ---------------------------------------------------------------------------
// BN statistics: stats[c] = sum_i y[i][c], stats[128+c] = sum_i y[i][c]^2
// ---------------------------------------------------------------------------
__global__ __launch_bounds__(256) void stats_kernel(
    const float* __restrict__ y, float* __restrict__ stats, int n)
{
  int c    = threadIdx.x & 127;
  int half = threadIdx.x >> 7;   // 0 or 1
  float s = 0.0f, sq = 0.0f;
  for (int r = blockIdx.x * 2 + half; r < n; r += gridDim.x * 2) {
    float v = y[(size_t)r * HD + c];
    s += v; sq += v * v;
  }
  __shared__ float ls[256], lq[256];
  ls[threadIdx.x] = s; lq[threadIdx.x] = sq;
  __syncthreads();
  if (half == 0) {
    s  += ls[128 + c];
    sq += lq[128 + c];
    unsafeAtomicAdd(&stats[c], s);
    unsafeAtomicAdd(&stats[HD + c], sq);
  }
}

// Fold BN into per-channel affine: ss[c] = gamma*rsqrt(var+eps), ss[128+c] = beta - mean*a
// (conv bias b cancels exactly in BN, so it is omitted everywhere)
__global__ void bnparam_kernel(const float* __restrict__ stats,
                               const float* __restrict__ gamma,
                               const float* __restrict__ beta,
                               float* __restrict__ ss, float invN)
{
  int c = threadIdx.x;
  if (c < HD) {
    float mean = stats[c] * invN;
    float var  = stats[HD + c] * invN - mean * mean;
    var = var < 0.0f ? 0.0f : var;
    float a = gamma[c] * rsqrtf(var + BN_EPS);
    ss[c]      = a;
    ss[HD + c] = beta[c] - mean * a;
  }
}

// ---------------------------------------------------------------------------
// Segment pooling: pooled[batch[i]] += leakyrelu(y[i]*a + s)
// ---------------------------------------------------------------------------
__global__ __launch_bounds__(256) void pool_kernel(
    const float* __restrict__ y, const float* __restrict__ ss,
    const int* __restrict__ batch, float* __restrict__ pooled, int n)
{
  long long t = (long long)blockIdx.x * blockDim.x + threadIdx.x;
  int node = (int)(t >> 5);
  int lane = (int)(t & 31);
  if (node >= n) return;
  int g  = batch[node];
  int c0 = lane * 4;
  float4 v = ((const float4*)(y + (size_t)node * HD))[lane];
  float vv[4] = {v.x, v.y, v.z, v.w};
#pragma unroll
  for (int j = 0; j < 4; ++j) {
    float z = vv[j] * ss[c0 + j] + ss[HD + c0 + j];
    z = z > 0.0f ? z : LEAKY * z;
    unsafeAtomicAdd(&pooled[g * HD + c0 + j], z);
  }
}

// out[g] = (pooled[g]/cnt[g]) @ Wl + bl    [G,10]
__global__ void head_kernel(const float* __restrict__ pooled,
                            const float* __restrict__ cnt,
                            const float* __restrict__ Wl,
                            const float* __restrict__ bl,
                            float* __restrict__ out, int G)
{
  int g = blockIdx.x * blockDim.x + threadIdx.x;
  if (g >= G) return;
  float inv = 1.0f / fmaxf(cnt[g], 1.0f);
  float acc[10];
#pragma unroll
  for (int c = 0; c < 10; ++c) acc[c] = bl[c];
  for (int k = 0; k < HD; ++k) {
    float p = pooled[g * HD + k] * inv;
#pragma unroll
    for (int c = 0; c < 10; ++c) acc[c] += p * Wl[k * 10 + c];
  }
#pragma unroll
  for (int c = 0; c < 10; ++c) out[g * 10 + c] = acc[c];
}

// ---------------------------------------------------------------------------
extern "C" void kernel_launch(void* const* d_in, const int* in_sizes, int n_in,
                              void* d_out, int out_size, void* d_ws, size_t ws_size,
                              hipStream_t stream)
{
  const float* x    = (const float*)d_in[0];
  const int*   eidx = (const int*)d_in[1];
  const int*   batc = (const int*)d_in[2];
  const float* W1   = (const float*)d_in[3];
  const float* g1   = (const float*)d_in[5];
  const float* be1  = (const float*)d_in[6];
  const float* W2   = (const float*)d_in[7];
  const float* g2   = (const float*)d_in[9];
  const float* be2  = (const float*)d_in[10];
  const float* W3   = (const float*)d_in[11];
  const float* g3   = (const float*)d_in[13];
  const float* be3  = (const float*)d_in[14];
  const float* Wl   = (const float*)d_in[15];
  const float* bl   = (const float*)d_in[16];

  int N = in_sizes[0] / HD;
  int E = in_sizes[1] / 2;
  int G = out_size / 10;
  const int* src = eidx;
  const int* dst = eidx + E;

  // workspace layout
  size_t off = 0;
  auto alloc = [&](size_t bytes) -> float* {
    float* p = (float*)((char*)d_ws + off);
    off += (bytes + 255) & ~(size_t)255;
    return p;
  };
  float* h      = alloc((size_t)N * HD * 4);
  float* y      = alloc((size_t)N * HD * 4);
  float* dis    = alloc((size_t)N * 4);
  float* stats  = alloc(2 * HD * 4);
  float* ss     = alloc(2 * HD * 4);
  float* pooled = alloc((size_t)G * HD * 4);
  float* cnt    = alloc((size_t)G * 4);

  float invN = 1.0f / (float)N;
  int gemmBlocks = (N + 63) / 64;
  long long aggThreads = (long long)(E + N) * 32;
  int aggBlocks  = (int)((aggThreads + 255) / 256);
  int poolBlocks = (int)(((long long)N * 32 + 255) / 256);

  // topology-only precompute
  hipMemsetAsync(dis, 0, (size_t)N * 4, stream);
  hipMemsetAsync(cnt, 0, (size_t)G * 4, stream);
  deg_kernel<<<(E + 255) / 256, 256, 0, stream>>>(dst, dis, E);
  dis_kernel<<<(N + 255) / 256, 256, 0, stream>>>(dis, N);
  cnt_kernel<<<(N + 255) / 256, 256, 0, stream>>>(batc, cnt, N);

  // ---- layer 1 ----
  gcn_gemm_kernel<<<gemmBlocks, 128, 0, stream>>>(x, W1, nullptr, h, N);
  hipMemsetAsync(y, 0, (size_t)N * HD * 4, stream);
  agg_kernel<<<aggBlocks, 256, 0, stream>>>(h, src, dst, dis, y, E, N);
  hipMemsetAsync(stats, 0, 2 * HD * 4, stream);
  stats_kernel<<<256, 256, 0, stream>>>(y, stats, N);
  bnparam_kernel<<<1, 128, 0, stream>>>(stats, g1, be1, ss, invN);

  // ---- layer 2 ----  (BN+lrelu of layer1 fused into A-load)
  gcn_gemm_kernel<<<gemmBlocks, 128, 0, stream>>>(y, W2, ss, h, N);
  hipMemsetAsync(y, 0, (size_t)N * HD * 4, stream);
  agg_kernel<<<aggBlocks, 256, 0, stream>>>(h, src, dst, dis, y, E, N);
  hipMemsetAsync(stats, 0, 2 * HD * 4, stream);
  stats_kernel<<<256, 256, 0, stream>>>(y, stats, N);
  bnparam_kernel<<<1, 128, 0, stream>>>(stats, g2, be2, ss, invN);

  // ---- layer 3 ----
  gcn_gemm_kernel<<<gemmBlocks, 128, 0, stream>>>(y, W3, ss, h, N);
  hipMemsetAsync(y, 0, (size_t)N * HD * 4, stream);
  agg_kernel<<<aggBlocks, 256, 0, stream>>>(h, src, dst, dis, y, E, N);
  hipMemsetAsync(stats, 0, 2 * HD * 4, stream);
  stats_kernel<<<256, 256, 0, stream>>>(y, stats, N);
  bnparam_kernel<<<1, 128, 0, stream>>>(stats, g3, be3, ss, invN);

  // ---- pool + head ----
  hipMemsetAsync(pooled, 0, (size_t)G * HD * 4, stream);
  pool_kernel<<<poolBlocks, 256, 0, stream>>>(y, ss, batc, pooled, N);
  head_kernel<<<(G + 127) / 128, 128, 0, stream>>>(pooled, cnt, Wl, bl, (float*)d_out, G);
}